// Assign_18219251269979
// MI455X (gfx1250) — compile-verified
//
#include <hip/hip_runtime.h>

typedef __attribute__((ext_vector_type(2))) float v2f;
typedef __attribute__((ext_vector_type(8))) float v8f;

#define DD 64
#define LDS_STRIDE 68   // 64 + 4 pad: b64 B-fragment reads hit 64 distinct LDS banks

// -------- mask helpers (mask[row] = 1 if row will be overwritten by scatter) -------
__global__ void clear_mask16(uint4* __restrict__ m, int n16) {
    int i = blockIdx.x * blockDim.x + threadIdx.x;
    if (i < n16) {
        uint4 z; z.x = 0u; z.y = 0u; z.z = 0u; z.w = 0u;
        m[i] = z;
    }
}

__global__ void set_mask(const int* __restrict__ tidx, unsigned char* __restrict__ mask, int k) {
    int i = blockIdx.x * blockDim.x + threadIdx.x;
    if (i < k) mask[tidx[i]] = 1;
}

// -------- copy x -> out, skipping rows that the scatter will overwrite -------------
__global__ void copy_rows(const float4* __restrict__ x, float4* __restrict__ out,
                          const unsigned char* __restrict__ mask, long long n4) {
    long long i = (long long)blockIdx.x * blockDim.x + threadIdx.x;
    if (i >= n4) return;
    if (mask) {
        long long row = i >> 4;          // 16 float4 per 64-float row
        if (mask[row]) return;
    }
    out[i] = x[i];
}

// -------- fused gather -> (16x64)=(16x64)x(64x64) WMMA GEMM -> bias/ReLU -> scatter
// One wave handles 16 gathered rows. f32 WMMA 16x16x4, K-dim 64 = 16 steps,
// 4 column blocks of 16 -> 64 v_wmma_f32_16x16x4_f32 per wave.
__global__ void gather_gemm_scatter(const float* __restrict__ x,
                                    const float* __restrict__ W,
                                    const float* __restrict__ bias,
                                    const int* __restrict__ arg_idx,
                                    const int* __restrict__ target_idx,
                                    float* __restrict__ out,
                                    int K_rows) {
    __shared__ float sWt[DD * LDS_STRIDE];   // W transposed: sWt[n*LDS_STRIDE + k] = W[k][n]

    // Cooperative transposed stage of W (row-major [k][n], 4096 floats) into LDS.
    for (int t = threadIdx.x; t < (DD * DD) / 4; t += blockDim.x) {
        int lin = t * 4;
        int k = lin >> 6;        // row of W
        int n = lin & 63;        // starting col
        float4 w4 = ((const float4*)W)[t];
        sWt[(n + 0) * LDS_STRIDE + k] = w4.x;
        sWt[(n + 1) * LDS_STRIDE + k] = w4.y;
        sWt[(n + 2) * LDS_STRIDE + k] = w4.z;
        sWt[(n + 3) * LDS_STRIDE + k] = w4.w;
    }
    __syncthreads();

    const int lane    = threadIdx.x & 31;
    const int wave    = threadIdx.x >> 5;
    const int tile    = blockIdx.x * (blockDim.x >> 5) + wave;
    const int row0    = tile * 16;
    if (row0 >= K_rows) return;           // wave-uniform: EXEC stays all-ones for WMMA

    const int halfSel = lane >> 4;        // 0 = lanes 0..15, 1 = lanes 16..31
    const int lm      = lane & 15;

    // Gathered source row for this lane's A-fragment (A 16x4 f32 layout:
    // lanes 0-15 hold K=0,1 ; lanes 16-31 hold K=2,3 of row M=lane%16).
    int m  = row0 + lm;
    int mc = (m < K_rows) ? m : (K_rows - 1);          // clamp for tail tiles
    const float* __restrict__ src = x + (long long)arg_idx[mc] * DD + halfSel * 2;

    // Bias per output column block (column N = nb*16 + lm in C/D layout).
    float bv0 = bias[ 0 + lm];
    float bv1 = bias[16 + lm];
    float bv2 = bias[32 + lm];
    float bv3 = bias[48 + lm];

    v8f acc0 = {}, acc1 = {}, acc2 = {}, acc3 = {};
    const float* __restrict__ wt = &sWt[lm * LDS_STRIDE + halfSel * 2];

#pragma unroll
    for (int kb = 0; kb < 16; ++kb) {
        v2f a  = *(const v2f*)(src + kb * 4);                               // global_load_b64
        v2f b0 = *(const v2f*)(wt + ( 0 * LDS_STRIDE) + kb * 4);            // ds_load_b64
        v2f b1 = *(const v2f*)(wt + (16 * LDS_STRIDE) + kb * 4);
        v2f b2 = *(const v2f*)(wt + (32 * LDS_STRIDE) + kb * 4);
        v2f b3 = *(const v2f*)(wt + (48 * LDS_STRIDE) + kb * 4);
        acc0 = __builtin_amdgcn_wmma_f32_16x16x4_f32(false, a, false, b0, (short)0, acc0, false, false);
        acc1 = __builtin_amdgcn_wmma_f32_16x16x4_f32(false, a, false, b1, (short)0, acc1, false, false);
        acc2 = __builtin_amdgcn_wmma_f32_16x16x4_f32(false, a, false, b2, (short)0, acc2, false, false);
        acc3 = __builtin_amdgcn_wmma_f32_16x16x4_f32(false, a, false, b3, (short)0, acc3, false, false);
    }

    // C/D layout: VGPR v, lanes 0-15 -> M=v, N=lane ; lanes 16-31 -> M=v+8, N=lane-16.
#pragma unroll
    for (int v = 0; v < 8; ++v) {
        int mo = row0 + v + halfSel * 8;
        if (mo < K_rows) {
            long long orow = (long long)target_idx[mo];
            float* dst = out + orow * DD + lm;
            float r0 = acc0[v] + bv0; dst[ 0] = r0 > 0.f ? r0 : 0.f;
            float r1 = acc1[v] + bv1; dst[16] = r1 > 0.f ? r1 : 0.f;
            float r2 = acc2[v] + bv2; dst[32] = r2 > 0.f ? r2 : 0.f;
            float r3 = acc3[v] + bv3; dst[48] = r3 > 0.f ? r3 : 0.f;
        }
    }
}

extern "C" void kernel_launch(void* const* d_in, const int* in_sizes, int n_in,
                              void* d_out, int out_size, void* d_ws, size_t ws_size,
                              hipStream_t stream) {
    const float* x          = (const float*)d_in[0];
    const float* W          = (const float*)d_in[1];
    const float* b          = (const float*)d_in[2];
    const int*   arg_idx    = (const int*)d_in[3];
    const int*   target_idx = (const int*)d_in[4];
    float*       out        = (float*)d_out;

    const int N = in_sizes[0] / DD;        // table rows
    const int K = in_sizes[3];             // gathered / scattered rows

    // Per-row overwrite mask in workspace (falls back to full copy if ws too small).
    const int n16 = (N + 15) / 16;         // uint4 granules for the clear
    unsigned char* mask = (ws_size >= (size_t)n16 * 16) ? (unsigned char*)d_ws : nullptr;

    if (mask) {
        clear_mask16<<<(n16 + 255) / 256, 256, 0, stream>>>((uint4*)mask, n16);
        set_mask<<<(K + 255) / 256, 256, 0, stream>>>(target_idx, mask, K);
    }

    const long long n4 = (long long)N * (DD / 4);
    copy_rows<<<(unsigned)((n4 + 255) / 256), 256, 0, stream>>>(
        (const float4*)x, (float4*)out, mask, n4);

    const int tiles = (K + 15) / 16;       // 16 gathered rows per wave
    const int wavesPerBlock = 8;           // 256 threads = 8 wave32
    const int blocks = (tiles + wavesPerBlock - 1) / wavesPerBlock;
    gather_gemm_scatter<<<blocks, 32 * wavesPerBlock, 0, stream>>>(
        x, W, b, arg_idx, target_idx, out, K);
}